// GNNStack_36928128811710
// MI455X (gfx1250) — compile-verified
//
#include <hip/hip_runtime.h>

#define NN 100000
#define EE 1600000
#define D  128
#define HM 256

typedef __attribute__((ext_vector_type(2))) float v2f;
typedef __attribute__((ext_vector_type(8))) float v8f;

// ---------------------------------------------------------------------------
// init: zero deg + agg, seed pred with bm2
// ---------------------------------------------------------------------------
__global__ void init_kernel(float* __restrict__ agg, float* __restrict__ deg,
                            float* __restrict__ pred, const float* __restrict__ bm2) {
  int i = blockIdx.x * blockDim.x + threadIdx.x;
  if (i < NN * D) agg[i] = 0.0f;
  if (i < NN) { deg[i] = 0.0f; pred[i] = bm2[0]; }
}

__global__ void zero_agg_kernel(float* __restrict__ agg) {
  int i = blockIdx.x * blockDim.x + threadIdx.x;
  if (i < NN * D) agg[i] = 0.0f;
}

// ---------------------------------------------------------------------------
// degree: deg[dst] += 1  (same for both layers, computed once)
// ---------------------------------------------------------------------------
__global__ void degree_kernel(const int* __restrict__ ei, float* __restrict__ deg) {
  int e = blockIdx.x * blockDim.x + threadIdx.x;
  if (e >= EE) return;
  atomicAdd(&deg[ei[EE + e]], 1.0f);
}

// ---------------------------------------------------------------------------
// scatter: agg[dst] += feat[src]; one wave32 per edge, float4 per lane
// (x fits in 192MB L2, so these gathers/atomics stay on-chip)
// ---------------------------------------------------------------------------
__global__ void scatter_kernel(const float* __restrict__ feat, const int* __restrict__ ei,
                               float* __restrict__ agg) {
  int idx  = blockIdx.x * blockDim.x + threadIdx.x;
  int e    = idx >> 5;
  int lane = idx & 31;
  int s = ei[e];
  int d = ei[EE + e];
  float4 v = *((const float4*)(feat + (size_t)s * D) + lane);
  float* o = agg + (size_t)d * D + lane * 4;
  atomicAdd(o + 0, v.x);
  atomicAdd(o + 1, v.y);
  atomicAdd(o + 2, v.z);
  atomicAdd(o + 3, v.w);
}

// ---------------------------------------------------------------------------
// SAGE combine: out = act( (agg/max(deg,1)) @ Wl + b + xin @ Wr )
// One wave per 16x16 tile; f32 WMMA, K-step 4; two accumulators to break the
// WMMA->WMMA RAW hazard chain.
// A-frag layout (32-bit 16x4): lanes 0-15 -> M=lane, K={kb,kb+1};
//                              lanes 16-31 -> M=lane-16, K={kb+2,kb+3}.
// C/D layout: VGPR g -> rows m0+g (lanes 0-15) / m0+g+8 (lanes 16-31), N=lane&15.
// ---------------------------------------------------------------------------
__global__ void sage_combine_kernel(const float* __restrict__ xin,
                                    const float* __restrict__ agg,
                                    const float* __restrict__ deg,
                                    const float* __restrict__ Wl,
                                    const float* __restrict__ bias,
                                    const float* __restrict__ Wr,
                                    float* __restrict__ out, int do_relu) {
  int wave = (blockIdx.x * blockDim.x + threadIdx.x) >> 5;
  int lane = threadIdx.x & 31;
  int tm = wave >> 3;            // 6250 row tiles
  int tn = wave & 7;             // 8 col tiles
  int m0 = tm << 4, n0 = tn << 4;
  int rl = lane & 15;
  int hi = lane >> 4;            // 0: K lo pair, 1: K hi pair
  int arow = m0 + rl;

  float rdeg = 1.0f / fmaxf(deg[arow], 1.0f);
  const float* xp = xin + (size_t)arow * D + hi * 2;
  const float* gp = agg + (size_t)arow * D + hi * 2;

  v8f accL = {};
  v8f accR = {};
#pragma unroll 4
  for (int kb = 0; kb < D; kb += 4) {
    v2f am; am[0] = gp[kb] * rdeg; am[1] = gp[kb + 1] * rdeg;
    v2f ax; ax[0] = xp[kb];        ax[1] = xp[kb + 1];
    const float* blp = Wl + (size_t)(kb + hi * 2) * D + n0 + rl;
    const float* brp = Wr + (size_t)(kb + hi * 2) * D + n0 + rl;
    v2f bl; bl[0] = blp[0]; bl[1] = blp[D];
    v2f br; br[0] = brp[0]; br[1] = brp[D];
    accL = __builtin_amdgcn_wmma_f32_16x16x4_f32(false, am, false, bl, (short)0, accL, false, false);
    accR = __builtin_amdgcn_wmma_f32_16x16x4_f32(false, ax, false, br, (short)0, accR, false, false);
  }

  float bv = bias[n0 + rl];
#pragma unroll
  for (int g = 0; g < 8; ++g) {
    float v = accL[g] + accR[g] + bv;
    if (do_relu) v = fmaxf(v, 0.0f);
    out[(size_t)(m0 + g + hi * 8) * D + n0 + rl] = v;
  }
}

// ---------------------------------------------------------------------------
// Fused MLP head: h1 = relu(relu(emb) @ Wm1 + bm1); pred += h1 @ Wm2
// One wave per (16-row, 16-hidden-col) tile; the 256->1 GEMM collapses to a
// per-lane scale + wave32 shfl_xor reduction + one atomic per row-half.
// ---------------------------------------------------------------------------
__global__ void mlp_kernel(const float* __restrict__ emb,
                           const float* __restrict__ Wm1,
                           const float* __restrict__ bm1,
                           const float* __restrict__ Wm2,
                           float* __restrict__ pred) {
  int wave = (blockIdx.x * blockDim.x + threadIdx.x) >> 5;
  int lane = threadIdx.x & 31;
  int tm = wave >> 4;            // 6250 row tiles
  int tn = wave & 15;            // 16 hidden-col tiles
  int m0 = tm << 4, n0 = tn << 4;
  int rl = lane & 15;
  int hi = lane >> 4;
  int arow = m0 + rl;

  const float* ep = emb + (size_t)arow * D + hi * 2;

  v8f accA = {};
  v8f accB = {};
#pragma unroll 4
  for (int kb = 0; kb < D; kb += 8) {
    v2f a0; a0[0] = fmaxf(ep[kb], 0.0f);     a0[1] = fmaxf(ep[kb + 1], 0.0f);
    v2f a1; a1[0] = fmaxf(ep[kb + 4], 0.0f); a1[1] = fmaxf(ep[kb + 5], 0.0f);
    const float* bp0 = Wm1 + (size_t)(kb + hi * 2) * HM + n0 + rl;
    const float* bp1 = Wm1 + (size_t)(kb + 4 + hi * 2) * HM + n0 + rl;
    v2f b0; b0[0] = bp0[0]; b0[1] = bp0[HM];
    v2f b1; b1[0] = bp1[0]; b1[1] = bp1[HM];
    accA = __builtin_amdgcn_wmma_f32_16x16x4_f32(false, a0, false, b0, (short)0, accA, false, false);
    accB = __builtin_amdgcn_wmma_f32_16x16x4_f32(false, a1, false, b1, (short)0, accB, false, false);
  }

  float b1v = bm1[n0 + rl];
  float w2  = Wm2[n0 + rl];
#pragma unroll
  for (int g = 0; g < 8; ++g) {
    float h1 = fmaxf(accA[g] + accB[g] + b1v, 0.0f);
    float c = h1 * w2;
    c += __shfl_xor(c, 8, 32);   // reduce within each 16-lane half
    c += __shfl_xor(c, 4, 32);
    c += __shfl_xor(c, 2, 32);
    c += __shfl_xor(c, 1, 32);
    if (rl == 0) atomicAdd(&pred[m0 + g + hi * 8], c);
  }
}

// ---------------------------------------------------------------------------
extern "C" void kernel_launch(void* const* d_in, const int* in_sizes, int n_in,
                              void* d_out, int out_size, void* d_ws, size_t ws_size,
                              hipStream_t stream) {
  const float* x    = (const float*)d_in[0];
  const int*   ei   = (const int*)  d_in[1];   // [2, E]: row0=src, row1=dst
  const float* Wl0  = (const float*)d_in[2];
  const float* b0   = (const float*)d_in[3];
  const float* Wr0  = (const float*)d_in[4];
  const float* Wl1  = (const float*)d_in[5];
  const float* b1   = (const float*)d_in[6];
  const float* Wr1  = (const float*)d_in[7];
  const float* Wm1  = (const float*)d_in[8];
  const float* bm1  = (const float*)d_in[9];
  const float* Wm2  = (const float*)d_in[10];
  const float* bm2  = (const float*)d_in[11];

  float* emb  = (float*)d_out;            // [N, 128]
  float* pred = emb + (size_t)NN * D;     // [N, 1]

  float* ws  = (float*)d_ws;
  float* deg = ws;                        // [N]
  float* agg = ws + NN;                   // [N, 128]
  float* h0  = agg + (size_t)NN * D;      // [N, 128]

  const int T = 256;
  const int gInit    = (NN * D + T - 1) / T;   // 50000
  const int gDeg     = (EE + T - 1) / T;       // 6250
  const int gScatter = (EE * 32) / T;          // 200000 (exact)
  const int gCombine = ((NN / 16) * 8 * 32) / T;   // 6250 (exact)
  const int gMlp     = ((NN / 16) * 16 * 32) / T;  // 12500 (exact)

  // deg + agg zero, pred = bm2
  init_kernel<<<gInit, T, 0, stream>>>(agg, deg, pred, bm2);
  degree_kernel<<<gDeg, T, 0, stream>>>(ei, deg);

  // layer 0
  scatter_kernel<<<gScatter, T, 0, stream>>>(x, ei, agg);
  sage_combine_kernel<<<gCombine, T, 0, stream>>>(x, agg, deg, Wl0, b0, Wr0, h0, 1);

  // layer 1
  zero_agg_kernel<<<gInit, T, 0, stream>>>(agg);
  scatter_kernel<<<gScatter, T, 0, stream>>>(h0, ei, agg);
  sage_combine_kernel<<<gCombine, T, 0, stream>>>(h0, agg, deg, Wl1, b1, Wr1, emb, 0);

  // post-MP head (reads emb, applies relu on load)
  mlp_kernel<<<gMlp, T, 0, stream>>>(emb, Wm1, bm1, Wm2, pred);
}